// MultiScaleRoIOperation_74783970558595
// MI455X (gfx1250) — compile-verified
//
#include <hip/hip_runtime.h>

// MI455X / gfx1250 multi-scale RoIAlign using V_WMMA_F32_16X16X4_F32.
//
// Separable formulation: out = Wy · F · Wx^T  per (roi, 16-channel block),
// where Wy [7x28] / Wx [7x28] are block-diagonal bilinear+bin-average weight
// matrices over the 28 (sample,corner) contributions per axis, and F is the
// gathered feature patch. Both stages run on the fp32 WMMA pipe (16x16x4).
//
// Features (109 MB total) are L2-resident (192 MB L2), so gathers amortize;
// HBM floor is ~160 MB of unique traffic (~7 us at 23.3 TB/s). Inner-loop
// structure: batch all 14 B-operand gathers per tile (uniform-base SADDR
// addressing, 32-bit offsets), then issue the 7 WMMAs back-to-back so load
// latency overlaps the matrix pipe instead of draining loadcnt per K-step.

#include <stdint.h>

typedef __attribute__((ext_vector_type(2))) float v2f;
typedef __attribute__((ext_vector_type(8))) float v8f;

#define ROI_C   256
#define ROI_OUT 7
#define NPB     512      // rois per batch image

__global__ __launch_bounds__(224)
void roi_align_wmma_kernel(const float* __restrict__ f0,
                           const float* __restrict__ f1,
                           const float* __restrict__ f2,
                           const float* __restrict__ f3,
                           const float* __restrict__ rois,
                           float* __restrict__ out)
{
    const int roi  = blockIdx.x;
    const int tid  = threadIdx.x;
    const int lane = tid & 31;
    const int wave = tid >> 5;       // 0..6
    const int m    = lane & 15;      // M / N index within wave half
    const int hi   = lane >> 4;      // lane half selects K pair (0,1) vs (2,3)

    // Per-axis contribution lists: j = 2*g + corner, g = sample index (0..13).
    __shared__ float s_wy[32];       // Wy weight per contribution (28 used)
    __shared__ int   s_oy[32];       // row*W element offset per contribution
    __shared__ float s_wx[32];
    __shared__ int   s_ox[32];       // col element offset per contribution
    __shared__ float sT[32][8][16];  // T[i][ph][c] stage-A -> stage-B tile (16 KB)

    // ---------------- per-RoI setup (redundant per thread, cheap) ------------
    const float x1 = rois[roi * 4 + 0];
    const float y1 = rois[roi * 4 + 1];
    const float x2 = rois[roi * 4 + 2];
    const float y2 = rois[roi * 4 + 3];

    // FPN level routing: k = floor(4 + log2(sqrt(area)/224)), clamp [2,5].
    const float area = (x2 - x1) * (y2 - y1);
    int lvl = (int)floorf(4.0f + log2f(sqrtf(area) * (1.0f / 224.0f)));
    lvl = min(max(lvl, 2), 5);
    const int   li    = lvl - 2;                 // 0..3
    const int   H     = 200 >> li;               // 200,100,50,25
    const int   W     = H;
    const int   HW    = H * W;
    const float scale = 1.0f / (float)(4 << li); // 2^-(li+2)

    const float* fsel  = (li == 0) ? f0 : (li == 1) ? f1 : (li == 2) ? f2 : f3;
    const int    bidx  = roi / NPB;
    const float* fbase = fsel + (size_t)bidx * ROI_C * HW;  // uniform -> SADDR

    const float x1s = x1 * scale, y1s = y1 * scale;
    const float x2s = x2 * scale, y2s = y2 * scale;
    const float rw = fmaxf(x2s - x1s, 1.0f);
    const float rh = fmaxf(y2s - y1s, 1.0f);

    if (tid < 32) { s_wy[tid] = 0.0f; s_wx[tid] = 0.0f; s_oy[tid] = 0; s_ox[tid] = 0; }
    __syncthreads();

    // Sample grid: coordinate = start + (g+0.5)*extent/14; torchvision semantics:
    // out-of-range (< -1 or > size) -> weight 0, else clamp to [0, size-1].
    // Fold the 0.5 bin-average factor and validity into the per-axis weights.
    if (tid < 14) {                       // y axis, wave 0
        const int   g  = tid;
        const float c  = y1s + ((float)g + 0.5f) * rh * (1.0f / 14.0f);
        const float vw = (c >= -1.0f && c <= (float)H) ? 0.5f : 0.0f;
        const float cc = fminf(fmaxf(c, 0.0f), (float)(H - 1));
        const float fl = floorf(cc);
        const int   i0 = (int)fl;
        const int   i1 = min(i0 + 1, H - 1);
        const float l  = cc - fl;
        s_oy[2 * g]     = i0 * W; s_wy[2 * g]     = (1.0f - l) * vw;
        s_oy[2 * g + 1] = i1 * W; s_wy[2 * g + 1] = l * vw;
    } else if (tid >= 32 && tid < 46) {   // x axis, wave 1
        const int   g  = tid - 32;
        const float c  = x1s + ((float)g + 0.5f) * rw * (1.0f / 14.0f);
        const float vw = (c >= -1.0f && c <= (float)W) ? 0.5f : 0.0f;
        const float cc = fminf(fmaxf(c, 0.0f), (float)(W - 1));
        const float fl = floorf(cc);
        const int   i0 = (int)fl;
        const int   i1 = min(i0 + 1, W - 1);
        const float l  = cc - fl;
        s_ox[2 * g]     = i0; s_wx[2 * g]     = (1.0f - l) * vw;
        s_ox[2 * g + 1] = i1; s_wx[2 * g + 1] = l * vw;
    }
    __syncthreads();

    // ---------------- hoist per-lane invariants into VGPRs -------------------
    // A-operand layout for 16x4 f32 (ISA 7.12.2): lanes 0-15 carry M with
    // VGPR0=K0,VGPR1=K1; lanes 16-31 carry M with VGPR0=K2,VGPR1=K3.
    // Wy/Wx are block-diagonal (row m touches only K-step k==m), so apply the
    // select on the *value* (v_cndmask), never on the LDS load.
    v2f aY[7], aX[7];
    uint32_t ro0[7], ro1[7];
    #pragma unroll
    for (int k = 0; k < 7; ++k) {
        const int jb = 4 * k + (hi ? 2 : 0);
        const float wy0 = s_wy[jb], wy1 = s_wy[jb + 1];
        const float wx0 = s_wx[jb], wx1 = s_wx[jb + 1];
        const bool  sel = (m == k);
        aY[k].x = sel ? wy0 : 0.0f;
        aY[k].y = sel ? wy1 : 0.0f;
        aX[k].x = sel ? wx0 : 0.0f;
        aX[k].y = sel ? wx1 : 0.0f;
        ro0[k] = (uint32_t)s_oy[jb];
        ro1[k] = (uint32_t)s_oy[jb + 1];
    }
    uint32_t col4[4];
    #pragma unroll
    for (int iv = 0; iv < 4; ++iv) col4[iv] = (uint32_t)s_ox[wave * 4 + iv];

    // ---------------- main loop over 16-channel blocks -----------------------
    for (int cb = 0; cb < ROI_C / 16; ++cb) {
        const int      c0   = cb * 16;
        const uint32_t coff = (uint32_t)(c0 + m) * (uint32_t)HW;  // lane channel

        // Stage A: T[ph, i, c] = sum_j Wy[ph,j] * F[row(j), col(i), c]
        // Wave w owns i in {4w .. 4w+3}; N-lane carries channel c0+m.
        #pragma unroll
        for (int iv = 0; iv < 4; ++iv) {
            const int      i    = wave * 4 + iv;
            const uint32_t base = coff + col4[iv];
            // Batch all 14 gathers (scalar base + 32-bit voffset), then WMMA.
            v2f bb[7];
            #pragma unroll
            for (int k = 0; k < 7; ++k) {
                bb[k].x = fbase[base + ro0[k]];
                bb[k].y = fbase[base + ro1[k]];
            }
            v8f acc = {};
            #pragma unroll
            for (int k = 0; k < 7; ++k)
                acc = __builtin_amdgcn_wmma_f32_16x16x4_f32(
                          false, aY[k], false, bb[k], (short)0, acc, false, false);
            // D layout: lane<16, VGPR r -> (M=r, N=lane). Rows M>=7 are 0.
            if (lane < 16) {
                #pragma unroll
                for (int r = 0; r < 8; ++r) sT[i][r][lane] = acc[r];
            }
        }
        __syncthreads();

        // Stage B: out[ph, pw, c] = sum_i Wx[pw,i] * T[ph, i, c]; wave w = ph.
        {
            const int ph = wave;
            v2f bb[7];
            #pragma unroll
            for (int k = 0; k < 7; ++k) {
                const int ib = 4 * k + (hi ? 2 : 0);
                bb[k].x = sT[ib][ph][m];
                bb[k].y = sT[ib + 1][ph][m];
            }
            v8f acc = {};
            #pragma unroll
            for (int k = 0; k < 7; ++k)
                acc = __builtin_amdgcn_wmma_f32_16x16x4_f32(
                          false, aX[k], false, bb[k], (short)0, acc, false, false);
            if (lane < 16) {
                const int c = c0 + lane;
                float* op = out + (((size_t)roi * ROI_C + c) * ROI_OUT + ph) * ROI_OUT;
                #pragma unroll
                for (int r = 0; r < ROI_OUT; ++r) op[r] = acc[r];
            }
        }
        __syncthreads();
    }
}

extern "C" void kernel_launch(void* const* d_in, const int* in_sizes, int n_in,
                              void* d_out, int out_size, void* d_ws, size_t ws_size,
                              hipStream_t stream) {
    const float* f0   = (const float*)d_in[0];
    const float* f1   = (const float*)d_in[1];
    const float* f2   = (const float*)d_in[2];
    const float* f3   = (const float*)d_in[3];
    const float* rois = (const float*)d_in[4];
    float*       out  = (float*)d_out;

    const int nroi = in_sizes[4] / 4;   // B*N = 1024
    roi_align_wmma_kernel<<<dim3(nroi), dim3(224), 0, stream>>>(
        f0, f1, f2, f3, rois, out);
}